// EdgeBlock_48258252538531
// MI455X (gfx1250) — compile-verified
//
#include <hip/hip_runtime.h>
#include <stdint.h>

typedef __attribute__((ext_vector_type(2))) float v2f;
typedef __attribute__((ext_vector_type(8))) float v8f;

#define N_NODES 100000
#define N_EDGES 1600000
#define C_IN 64
#define C_OUT 64

// Pitches chosen so pitch % 64 == 4 (floats): with 64 banks and b64 DS
// accesses, lane L touches banks {4m + 2*halfk, +1} -> disjoint across all
// 32 lanes (conflict-free for both halves of the wave).
#define W1_PITCH 132   // lW1[n][k], k = 0..127
#define W2_PITCH 68    // lW2[n][k], k = 0..63
#define H_PITCH  68    // per-wave h1 staging [32 rows][64 cols]

__global__ void zero_out_kernel(float* __restrict__ out, int n) {
    int i = blockIdx.x * blockDim.x + threadIdx.x;
    if (i < n) out[i] = 0.0f;
}

__global__ __launch_bounds__(256) void edgeconv_kernel(
    const float* __restrict__ x,
    const long long* __restrict__ edge_index,
    const float* __restrict__ W1, const float* __restrict__ b1,
    const float* __restrict__ W2, const float* __restrict__ b2,
    float* __restrict__ out)
{
    __shared__ __align__(16) float lW1[64 * W1_PITCH];
    __shared__ __align__(16) float lW2[64 * W2_PITCH];
    __shared__ __align__(16) float lH[8][32 * H_PITCH];

    const int tid = threadIdx.x;

    // Stage W1^T, W2^T into LDS: lW[n][k] = W[k][n]; B-fragment = one ds_load_b64.
    for (int i = tid; i < 128 * 64; i += 256) {
        int k = i >> 6, n = i & 63;
        lW1[n * W1_PITCH + k] = W1[i];
    }
    for (int i = tid; i < 64 * 64; i += 256) {
        int k = i >> 6, n = i & 63;
        lW2[n * W2_PITCH + k] = W2[i];
    }
    __syncthreads();

    const int wave  = tid >> 5;
    const int lane  = tid & 31;
    const int m     = lane & 15;   // A row within tile / B,D column
    const int halfk = lane >> 4;   // which K-pair this lane holds

    // Each wave owns 32 edges (two 16-row A tiles sharing the B fragments).
    // E = 1,600,000 = 6,250 blocks * 8 waves * 32 edges (no tail).
    const int base = (blockIdx.x * 8 + wave) * 32;
    const int e0 = base + m;
    const int e1 = base + 16 + m;
    const int src0 = (int)edge_index[e0];
    const int dst0 = (int)edge_index[N_EDGES + e0];
    const int src1 = (int)edge_index[e1];
    const int dst1 = (int)edge_index[N_EDGES + e1];

    const float2* xi0 = (const float2*)(x + (size_t)dst0 * C_IN);
    const float2* xj0 = (const float2*)(x + (size_t)src0 * C_IN);
    const float2* xi1 = (const float2*)(x + (size_t)dst1 * C_IN);
    const float2* xj1 = (const float2*)(x + (size_t)src1 * C_IN);

    v8f acc[2][4];
    #pragma unroll
    for (int t = 0; t < 2; ++t)
        #pragma unroll
        for (int n = 0; n < 4; ++n)
            acc[t][n] = (v8f){0.f,0.f,0.f,0.f,0.f,0.f,0.f,0.f};

    // ---------------- Layer 1: [2 x 16 x 128] @ W1[128 x 64] ----------------
    // Phase A: k = 0..63  -> h = xi   (no branch: loop split keeps it uniform)
    #pragma unroll 4
    for (int kk = 0; kk < 16; ++kk) {
        const int k0 = 4 * kk + 2 * halfk;             // even, 0..62
        float2 t0 = xi0[k0 >> 1];
        float2 t1 = xi1[k0 >> 1];
        v2f a0; a0.x = t0.x; a0.y = t0.y;
        v2f a1; a1.x = t1.x; a1.y = t1.y;
        const float* wb = lW1 + k0;
        #pragma unroll
        for (int n = 0; n < 4; ++n) {
            v2f b = *(const v2f*)(wb + (n * 16 + m) * W1_PITCH);
            acc[0][n] = __builtin_amdgcn_wmma_f32_16x16x4_f32(false, a0, false, b, (short)0, acc[0][n], false, false);
            acc[1][n] = __builtin_amdgcn_wmma_f32_16x16x4_f32(false, a1, false, b, (short)0, acc[1][n], false, false);
        }
    }
    // Phase B: k = 64..127 -> h = xj - xi
    #pragma unroll 4
    for (int kk = 0; kk < 16; ++kk) {
        const int k0 = 4 * kk + 2 * halfk;             // even, 0..62 (feature k0 in xi/xj)
        float2 ti0 = xi0[k0 >> 1], tj0 = xj0[k0 >> 1];
        float2 ti1 = xi1[k0 >> 1], tj1 = xj1[k0 >> 1];
        v2f a0; a0.x = tj0.x - ti0.x; a0.y = tj0.y - ti0.y;
        v2f a1; a1.x = tj1.x - ti1.x; a1.y = tj1.y - ti1.y;
        const float* wb = lW1 + 64 + k0;
        #pragma unroll
        for (int n = 0; n < 4; ++n) {
            v2f b = *(const v2f*)(wb + (n * 16 + m) * W1_PITCH);
            acc[0][n] = __builtin_amdgcn_wmma_f32_16x16x4_f32(false, a0, false, b, (short)0, acc[0][n], false, false);
            acc[1][n] = __builtin_amdgcn_wmma_f32_16x16x4_f32(false, a1, false, b, (short)0, acc[1][n], false, false);
        }
    }

    // Bias + ReLU; stage h1 (D layout -> row-major LDS) for layer-2 A-frags.
    float* hrow = lH[wave];
    {
        float bb[4];
        #pragma unroll
        for (int n = 0; n < 4; ++n) bb[n] = b1[n * 16 + m];
        #pragma unroll
        for (int t = 0; t < 2; ++t)
            #pragma unroll
            for (int v = 0; v < 8; ++v) {
                const int Mr = t * 16 + v + 8 * halfk;
                #pragma unroll
                for (int n = 0; n < 4; ++n)
                    hrow[Mr * H_PITCH + n * 16 + m] = fmaxf(acc[t][n][v] + bb[n], 0.0f);
            }
    }
    // Same-wave DS ops are in-order (DScnt); subsequent reads see these stores.

    // ---------------- Layer 2: [2 x 16 x 64] @ W2[64 x 64] ----------------
    #pragma unroll
    for (int t = 0; t < 2; ++t)
        #pragma unroll
        for (int n = 0; n < 4; ++n)
            acc[t][n] = (v8f){0.f,0.f,0.f,0.f,0.f,0.f,0.f,0.f};

    const float* hA0 = hrow + m * H_PITCH;
    const float* hA1 = hrow + (16 + m) * H_PITCH;
    #pragma unroll 4
    for (int kk = 0; kk < 16; ++kk) {
        const int k0 = 4 * kk + 2 * halfk;
        v2f a0 = *(const v2f*)(hA0 + k0);
        v2f a1 = *(const v2f*)(hA1 + k0);
        const float* wb = lW2 + k0;
        #pragma unroll
        for (int n = 0; n < 4; ++n) {
            v2f b = *(const v2f*)(wb + (n * 16 + m) * W2_PITCH);
            acc[0][n] = __builtin_amdgcn_wmma_f32_16x16x4_f32(false, a0, false, b, (short)0, acc[0][n], false, false);
            acc[1][n] = __builtin_amdgcn_wmma_f32_16x16x4_f32(false, a1, false, b, (short)0, acc[1][n], false, false);
        }
    }

    // Bias + ReLU + scatter-max. Values >= 0 and out is zero-initialized, so
    // signed-int atomic max on float bit patterns == float segment_max.
    {
        float bb[4];
        #pragma unroll
        for (int n = 0; n < 4; ++n) bb[n] = b2[n * 16 + m];
        #pragma unroll
        for (int t = 0; t < 2; ++t) {
            const int dstT = t ? dst1 : dst0;
            #pragma unroll
            for (int v = 0; v < 8; ++v) {
                const int Mr   = v + 8 * halfk;
                const int dstR = __shfl(dstT, Mr, 32);   // dst of edge row Mr of tile t
                int* prow = (int*)(out + (size_t)dstR * C_OUT);
                #pragma unroll
                for (int n = 0; n < 4; ++n)
                    atomicMax(prow + n * 16 + m,
                              __float_as_int(fmaxf(acc[t][n][v] + bb[n], 0.0f)));
            }
        }
    }
}

extern "C" void kernel_launch(void* const* d_in, const int* in_sizes, int n_in,
                              void* d_out, int out_size, void* d_ws, size_t ws_size,
                              hipStream_t stream) {
    const float*     x  = (const float*)d_in[0];
    const long long* ei = (const long long*)d_in[1];   // int64 edge_index [2, E]
    const float*     W1 = (const float*)d_in[2];
    const float*     b1 = (const float*)d_in[3];
    const float*     W2 = (const float*)d_in[4];
    const float*     b2 = (const float*)d_in[5];
    float* out = (float*)d_out;

    const int n_out = N_NODES * C_OUT;
    zero_out_kernel<<<(n_out + 255) / 256, 256, 0, stream>>>(out, n_out);

    // 6,250 blocks * 8 waves * 32 edges = 1,600,000 edges exactly.
    edgeconv_kernel<<<N_EDGES / 256, 256, 0, stream>>>(x, ei, W1, b1, W2, b2, out);
}